// YOLOPredict_62569083568617
// MI455X (gfx1250) — compile-verified
//
#include <hip/hip_runtime.h>
#include <hip/hip_bf16.h>
#include <stdint.h>

// ---------------- problem constants ----------------
#define SS        7
#define NB        2
#define NC        20
#define NBOX      98          // SS*SS*NB
#define NPAD      112         // 7 tiles of 16
#define NTILE     7           // NPAD/16
#define PRED_PER  1470        // SS*SS*(NB*5+NC) = 49*30
#define OUT_PER   11760       // NC*NBOX*6
#define CONF_TH   0.1f
#define IOU_TH    0.5f
#define NTHREADS  256
#define NWAVES    8

typedef __attribute__((ext_vector_type(2))) float    v2f;
typedef __attribute__((ext_vector_type(4))) float    v4f;
typedef __attribute__((ext_vector_type(8))) float    v8f;
typedef __attribute__((ext_vector_type(4))) unsigned uv4;
typedef __attribute__((ext_vector_type(4))) int      iv4;
typedef __attribute__((ext_vector_type(8))) int      iv8;

#if defined(__gfx1250__) && __has_builtin(__builtin_amdgcn_tensor_load_to_lds)
#define HAVE_TDM 1
#else
#define HAVE_TDM 0
#endif

// Definitive stderr probe for which staging path the DEVICE pass compiled.
#if defined(__gfx1250__)
#if HAVE_TDM
#if __clang_major__ >= 23
#pragma message("CDNA5-PROBE: device pass gfx1250, TDM path ENABLED, 6-arg builtin (clang>=23)")
#else
#pragma message("CDNA5-PROBE: device pass gfx1250, TDM path ENABLED, 5-arg builtin (clang<23)")
#endif
#else
#pragma message("CDNA5-PROBE: device pass gfx1250, TDM builtin ABSENT, fallback loads")
#endif
#endif

__device__ __forceinline__ float clamp01(float v) {
    return fminf(fmaxf(v, 0.0f), 1.0f);
}

// wave32 xor-swizzle reduction step (ds_swizzle_b32: and=0x1f, or=0, xor=X)
#define SWZ_F(v, X) __builtin_bit_cast(float, __builtin_amdgcn_ds_swizzle(__builtin_bit_cast(int, (v)), ((X) << 10) | 0x1f))
#define SWZ_I(v, X) __builtin_amdgcn_ds_swizzle((v), ((X) << 10) | 0x1f)
#define REDSTEP(X)                                                      \
    {                                                                   \
        float ov = SWZ_F(bestv, X);                                     \
        int   oj = SWZ_I(bestj, X);                                     \
        if (ov > bestv || (ov == bestv && oj < bestj)) {                \
            bestv = ov; bestj = oj;                                     \
        }                                                               \
    }

__global__ __launch_bounds__(NTHREADS)
void yolo_nms_kernel(const float* __restrict__ pred, float* __restrict__ out)
{
    const int b    = blockIdx.x;
    const int tid  = threadIdx.x;
    const int lane = tid & 31;
    const int wave = tid >> 5;

    __shared__ float    s_pred[PRED_PER];       // raw prediction for this image
    __shared__ float    s_box[NPAD][4];         // decoded clipped xyxy (pad = 0)
    __shared__ float    s_area[NPAD];           // box areas (pad = 0)
    __shared__ float    s_score[NC][NPAD];      // per-class scores (pad = 0)
    __shared__ __align__(16) unsigned s_sup[NPAD][4]; // row o: bit j => iou(o,j) > 0.5
    __shared__ float    s_keep[NC][NPAD];       // final keep flags (0/1)

    // ---------------- Phase A: stage pred into LDS ----------------
    const float* p = pred + (size_t)b * PRED_PER;
#if HAVE_TDM
    if (wave == 0) {   // uniform branch: one TDM issue for the block
        const unsigned long long ga = (unsigned long long)(uintptr_t)p;
        const unsigned lds_off = (unsigned)(uintptr_t)&s_pred[0]; // low 32 bits = LDS offset
        uv4 g0;
        g0.x = 1u;                                            // count=1, user mode
        g0.y = lds_off;                                       // lds_addr [63:32]
        g0.z = (unsigned)(ga & 0xFFFFFFFFull);                // global_addr lo
        g0.w = (unsigned)((ga >> 32) & 0x01FFFFFFull)         // global_addr [56:32]
             | (2u << 30);                                    // type=2 ("image") [127:126]
        iv8 g1;
        g1[0] = (int)(2u << 16);                              // wg_mask=0, data_size=4B
        g1[1] = (int)(((unsigned)PRED_PER & 0xFFFFu) << 16);  // tensor_dim0 lo16 @ [63:48]
        g1[2] = (int)(((unsigned)PRED_PER >> 16) | (1u << 16)); // dim0 hi16, tensor_dim1=1
        g1[3] = (int)((unsigned)PRED_PER << 16);              // tile_dim0=1470 @ [127:112]
        g1[4] = 1;                                            // tile_dim1=1, tile_dim2=0
        g1[5] = PRED_PER;                                     // tensor_dim0_stride lo32
        g1[6] = 0;
        g1[7] = 0;
        iv4 gz = {0, 0, 0, 0};
#if __clang_major__ >= 23
        iv8 gz8 = {0, 0, 0, 0, 0, 0, 0, 0};
        __builtin_amdgcn_tensor_load_to_lds(g0, g1, gz, gz, gz8, 0);
#else
        __builtin_amdgcn_tensor_load_to_lds(g0, g1, gz, gz, 0);
#endif
        __builtin_amdgcn_s_wait_tensorcnt(0);
    }
#else
    __builtin_prefetch(p, 0, 3);                              // global_prefetch_b8
    for (int i = tid; i < PRED_PER; i += NTHREADS)
        s_pred[i] = __builtin_nontemporal_load(&p[i]);
#endif
    // zero suppression bitmask
    for (int i = tid; i < NPAD * 4; i += NTHREADS) ((unsigned*)s_sup)[i] = 0u;
    __syncthreads();

    // ---------------- Phase B: decode boxes, areas, scores ----------------
    for (int n = tid; n < NPAD; n += NTHREADS) {
        float x0 = 0.f, y0 = 0.f, x1 = 0.f, y1 = 0.f, area = 0.f;
        if (n < NBOX) {
            const int cell = n >> 1;
            const int bi   = n & 1;
            const int ii   = cell / SS;
            const int jj   = cell - ii * SS;
            const float* q = &s_pred[cell * 30 + bi * 5];
            const float bx = q[0], by = q[1], bw = q[2], bh = q[3];
            const float cx = (bx + (float)ii) * (1.0f / (float)SS);
            const float cy = (by + (float)jj) * (1.0f / (float)SS);
            x0 = clamp01(cx - 0.5f * bw);
            y0 = clamp01(cy - 0.5f * bh);
            x1 = clamp01(cx + 0.5f * bw);
            y1 = clamp01(cy + 0.5f * bh);
            area = (x1 - x0) * (y1 - y0);
        }
        s_box[n][0] = x0; s_box[n][1] = y0;
        s_box[n][2] = x1; s_box[n][3] = y1;
        s_area[n]   = area;
    }
    for (int idx = tid; idx < NC * NPAD; idx += NTHREADS) {
        const int c = idx / NPAD;
        const int n = idx - c * NPAD;
        float sc = 0.f;
        if (n < NBOX) {
            const int cell = n >> 1;
            const int bi   = n & 1;
            sc = s_pred[cell * 30 + bi * 5 + 4] * s_pred[cell * 30 + 10 + c];
        }
        s_score[c][n] = sc;
    }
    __syncthreads();

    // ---------------- Phase C: suppression bitmask via WMMA ----------------
    // D = A(16x4) x B(4x16): A[m] = (area_m, 1, 0, 0), B[:,n] = (1, area_n, 0, 0)
    // => D[m][n] = area_m + area_n  (IoU denominator before -inter)
    // Branch-free operand build (padded boxes are all-zero => sup always false,
    // so no mm/nn < 98 guard is needed).
    const int   nloc = lane & 15;
    const int   hf   = lane >> 4;             // 0: lanes 0-15, 1: lanes 16-31
    const float m0   = (hf == 0) ? 1.0f : 0.0f;
    for (int t = wave; t < NTILE * NTILE; t += NWAVES) {
        const int ti = t / NTILE;
        const int tj = t - ti * NTILE;

        const float am = s_area[ti * 16 + nloc];
        const float an = s_area[tj * 16 + nloc];
        v2f A, Bv;
        A.x  = am * m0;  A.y  = m0;           // lanes 0-15: K0=area_m, K1=1; else 0
        Bv.x = m0;       Bv.y = an * m0;      // lanes 0-15: K0 row=1, K1 row=area_n
        v8f Cacc = {};
        v8f D = __builtin_amdgcn_wmma_f32_16x16x4_f32(
            false, A, false, Bv, (short)0, Cacc, false, false);

        const int nn = tj * 16 + nloc;
        const float bx0 = s_box[nn][0], by0 = s_box[nn][1];
        const float bx1 = s_box[nn][2], by1 = s_box[nn][3];
#pragma unroll
        for (int r = 0; r < 8; ++r) {
            const int mm = ti * 16 + r + hf * 8;   // D vgpr r: M=r (lanes0-15), M=r+8
            const float ltx = fmaxf(s_box[mm][0], bx0);
            const float lty = fmaxf(s_box[mm][1], by0);
            const float rbx = fminf(s_box[mm][2], bx1);
            const float rby = fminf(s_box[mm][3], by1);
            const float w = fmaxf(rbx - ltx, 0.0f);
            const float h = fmaxf(rby - lty, 0.0f);
            const float inter = w * h;
            const float uni   = D[r] - inter;      // area_m + area_n - inter
            const bool  sup   = inter > IOU_TH * uni;   // iou > 0.5
            const unsigned bal = __builtin_amdgcn_ballot_w32(sup);
            if (nloc == 0) {
                const unsigned bits16 = hf ? (bal >> 16) : (bal & 0xFFFFu);
                atomicOr(&s_sup[mm][tj >> 1], bits16 << ((tj & 1) * 16)); // ds_or_b32
            }
        }
    }
    __syncthreads();

    // ---------------- Phase D: greedy NMS, one wave per class ----------------
    for (int c = wave; c < NC; c += NWAVES) {
        // each lane owns slots j = lane + 32*s, s = 0..3
        unsigned keep = 0u, proc = 0u;
        float sc[4];
#pragma unroll
        for (int s = 0; s < 4; ++s) {
            const int j = lane + 32 * s;
            const float v = (j < NBOX) ? s_score[c][j] : 0.0f;
            sc[s] = v;
            if (j < NBOX && v > CONF_TH) keep |= (1u << s);
        }
        for (int iter = 0; iter < NBOX; ++iter) {
            // per-lane argmax over alive, unprocessed slots
            float bestv = -1.0f;
            int   bestj = 1 << 30;
            const unsigned cand = keep & ~proc;
#pragma unroll
            for (int s = 0; s < 4; ++s) {
                if (cand & (1u << s)) {
                    const int j = lane + 32 * s;
                    if (sc[s] > bestv || (sc[s] == bestv && j < bestj)) {
                        bestv = sc[s]; bestj = j;
                    }
                }
            }
            // wave32 reduction: max score, min index on ties (== stable argsort order)
            REDSTEP(16) REDSTEP(8) REDSTEP(4) REDSTEP(2) REDSTEP(1)
            if (bestj >= NPAD) break;   // nothing left to select
            const int o = bestj;
            if ((o & 31) == lane) proc |= 1u << (o >> 5);
            // one ds_load_b128 broadcast of row o, then per-lane bit tests
            const uv4 row = *((const uv4*)&s_sup[o][0]);
#pragma unroll
            for (int s = 0; s < 4; ++s) {
                const int j = lane + 32 * s;
                if (((row[s] >> lane) & 1u) && j != o) keep &= ~(1u << s);
            }
        }
#pragma unroll
        for (int s = 0; s < 4; ++s) {
            const int j = lane + 32 * s;
            if (j < NPAD) s_keep[c][j] = (keep >> s) & 1u ? 1.0f : 0.0f;
        }
    }
    __syncthreads();

    // ---------------- Phase E: coalesced nontemporal float4 output ----------------
    // out[b][c][n][f], f: 4 box coords * k, score * k, k   (OUT_PER % 4 == 0)
    v4f* ob4 = (v4f*)(out + (size_t)b * OUT_PER);
    for (int q = tid; q < OUT_PER / 4; q += NTHREADS) {
        v4f v;
#pragma unroll
        for (int k = 0; k < 4; ++k) {
            const int idx = q * 4 + k;
            const int c = idx / (NBOX * 6);
            const int r = idx - c * (NBOX * 6);
            const int n = r / 6;
            const int f = r - n * 6;
            const float kf = s_keep[c][n];
            float val;
            if (f < 4)       val = s_box[n][f] * kf;
            else if (f == 4) val = s_score[c][n] * kf;
            else             val = kf;
            v[k] = val;
        }
        __builtin_nontemporal_store(v, &ob4[q]);
    }
}

extern "C" void kernel_launch(void* const* d_in, const int* in_sizes, int n_in,
                              void* d_out, int out_size, void* d_ws, size_t ws_size,
                              hipStream_t stream) {
    const float* pred = (const float*)d_in[0];
    float* out = (float*)d_out;
    const int bs = in_sizes[0] / PRED_PER;   // 8192
    yolo_nms_kernel<<<dim3(bs), dim3(NTHREADS), 0, stream>>>(pred, out);
}